// HarmonicOscillatorSinCos_8280696947388
// MI455X (gfx1250) — compile-verified
//
#include <hip/hip_runtime.h>
#include <math.h>

#define FS_F     16000.0f
#define FR       80
#define T_FRAMES 1250
#define T_OUT    (T_FRAMES * FR)            // 100000
#define NHARM    50
#define BATCH    8
#define CHUNK    256
#define NC       ((T_OUT + CHUNK - 1) / CHUNK)   // 391
#define WIN      (2 * FR + 1)               // 161
#define TWO_PI   6.28318530717958647692f

typedef __attribute__((ext_vector_type(2))) float v2f;
typedef __attribute__((ext_vector_type(8))) float v8f;

// f0 linear interpolation (torch F.interpolate, align_corners=False), /FS folded by caller
__device__ __forceinline__ float f0_up_val(const float* __restrict__ f0b, int t) {
    float pos = ((float)t + 0.5f) * (1.0f / (float)FR) - 0.5f;
    pos = fminf(fmaxf(pos, 0.0f), (float)(T_FRAMES - 1));
    int i0 = (int)floorf(pos);
    int i1 = min(i0 + 1, T_FRAMES - 1);
    float w = pos - (float)i0;
    return f0b[i0] * (1.0f - w) + f0b[i1] * w;
}

__device__ __forceinline__ float hannw(int m) {
    // periodic hann of length 161: 0.5*(1 - cos(2*pi*m/161))
    return 0.5f * (1.0f - cosf((TWO_PI / (float)WIN) * (float)m));
}

// ---------------- Kernel A: per-chunk sums of f0_up/FS ----------------
__global__ __launch_bounds__(CHUNK) void chunk_sum_kernel(const float* __restrict__ f0,
                                                          float* __restrict__ chunkSum) {
    int blk = blockIdx.x;
    int b = blk / NC, c = blk % NC;
    int tid = threadIdx.x;
    int t = c * CHUNK + tid;
    const float* f0b = f0 + b * T_FRAMES;
    float x = (t < T_OUT) ? f0_up_val(f0b, t) * (1.0f / FS_F) : 0.0f;
    __shared__ float red[CHUNK];
    red[tid] = x;
    __syncthreads();
    for (int off = CHUNK / 2; off > 0; off >>= 1) {
        if (tid < off) red[tid] += red[tid + off];
        __syncthreads();
    }
    if (tid == 0) chunkSum[b * NC + c] = red[0];
}

// ---------------- Kernel B: per-batch exclusive scan of chunk sums ----------------
__global__ __launch_bounds__(512) void chunk_scan_kernel(const float* __restrict__ chunkSum,
                                                         float* __restrict__ chunkBase) {
    int b = blockIdx.x;
    int tid = threadIdx.x;
    __shared__ float s[512];
    float own = (tid < NC) ? chunkSum[b * NC + tid] : 0.0f;
    s[tid] = own;
    __syncthreads();
    for (int off = 1; off < 512; off <<= 1) {
        float add = (tid >= off) ? s[tid - off] : 0.0f;
        __syncthreads();
        s[tid] += add;
        __syncthreads();
    }
    if (tid < NC) chunkBase[b * NC + tid] = s[tid] - own;   // exclusive base
}

// ---------------- Kernel C: WMMA in-block scan + harmonic oscillator ----------------
__global__ __launch_bounds__(CHUNK) void osc_kernel(const float* __restrict__ f0,
                                                    const float* __restrict__ cn_sin,
                                                    const float* __restrict__ cn_cos,
                                                    const float* __restrict__ an_sin,
                                                    const float* __restrict__ an_cos,
                                                    const float* __restrict__ chunkBase,
                                                    float* __restrict__ out) {
    int blk = blockIdx.x;
    int b = blk / NC, c = blk % NC;
    int tid = threadIdx.x;
    int t0 = c * CHUNK;
    int t = t0 + tid;
    const float* f0b = f0 + b * T_FRAMES;

    __shared__ float sx[CHUNK];            // scan values, later scan result
    __shared__ float stmp[CHUNK];          // V*S intermediate (C-layout -> B-layout transpose)
    __shared__ float s_cnS[6 * NHARM];     // staged cn_sin frame columns [frame][harm]
    __shared__ float s_cnC[6 * NHARM];

    float f0u = 0.0f, x = 0.0f;
    if (t < T_OUT) { f0u = f0_up_val(f0b, t); x = f0u * (1.0f / FS_F); }
    sx[tid] = x;

    // stage 6 frame columns (F0..F0+5) of cn_sin/cn_cos; zero beyond tensor
    int F0 = t0 / FR;
    for (int l = tid; l < 2 * 6 * NHARM; l += CHUNK) {
        int which = (l >= 6 * NHARM) ? 1 : 0;
        int li = which ? (l - 6 * NHARM) : l;
        int f = li / NHARM;
        int n = li % NHARM;
        int gf = F0 + f;
        float v = 0.0f;
        if (gf < T_FRAMES) {
            const float* src = which ? cn_cos : cn_sin;
            v = src[(b * NHARM + n) * T_FRAMES + gf];
        }
        if (which) s_cnC[li] = v; else s_cnS[li] = v;
    }
    __syncthreads();

    // ---- wave 0: inclusive scan of sx[0..255] via WMMA (tensor-core scan) ----
    // Column-major tiling: V[i][j] = sx[j*16 + i].  R = Lt*V + J*(V*S)
    //   Lt[i][i'] = (i' <= i), S[j'][j] = (j' < j), J = all-ones.
    if (tid < 32) {
        int lane = tid;
        int half = (lane >= 16) ? 1 : 0;   // A/B layout: lanes 16-31 hold K/rows +2
        int q = lane & 15;                 // M (A) or N (B) index
        v8f acc = {};                      // Lt*V  (+ J*T1 later)
        v8f t1  = {};                      // V*S
        #pragma unroll
        for (int k = 0; k < 4; ++k) {
            int kb = 4 * k + 2 * half;
            // A slice of V (16x4): a.x=V[q][kb], a.y=V[q][kb+1]
            v2f av; av.x = sx[kb * 16 + q]; av.y = sx[(kb + 1) * 16 + q];
            // B slice of S (4x16): S[r][j] = r < j
            v2f bs; bs.x = (kb < q) ? 1.0f : 0.0f; bs.y = ((kb + 1) < q) ? 1.0f : 0.0f;
            t1 = __builtin_amdgcn_wmma_f32_16x16x4_f32(false, av, false, bs,
                                                       (short)0, t1, false, false);
            // A slice of Lt: Lt[i][i'] = i' <= i
            v2f al; al.x = (kb <= q) ? 1.0f : 0.0f; al.y = ((kb + 1) <= q) ? 1.0f : 0.0f;
            // B slice of V: b.x=V[kb][q], b.y=V[kb+1][q]
            v2f bv; bv.x = sx[q * 16 + kb]; bv.y = sx[q * 16 + kb + 1];
            acc = __builtin_amdgcn_wmma_f32_16x16x4_f32(false, al, false, bv,
                                                        (short)0, acc, false, false);
        }
        // transpose T1 (C/D layout) through LDS into B layout
        #pragma unroll
        for (int v = 0; v < 8; ++v) {
            int i = v + 8 * half;          // C/D layout: VGPR v, lanes 16-31 hold M+8
            stmp[i * 16 + q] = t1[v];
        }
        v2f aj; aj.x = 1.0f; aj.y = 1.0f;  // all-ones A slice of J
        #pragma unroll
        for (int k = 0; k < 4; ++k) {
            int kb = 4 * k + 2 * half;
            v2f bt; bt.x = stmp[kb * 16 + q]; bt.y = stmp[(kb + 1) * 16 + q];
            acc = __builtin_amdgcn_wmma_f32_16x16x4_f32(false, aj, false, bt,
                                                        (short)0, acc, false, false);
        }
        // write back: value index p = j*16 + i  (column-major tiling)
        #pragma unroll
        for (int v = 0; v < 8; ++v) {
            int i = v + 8 * half;
            sx[q * 16 + i] = acc[v];
        }
    }
    __syncthreads();

    if (t >= T_OUT) return;
    float Cval = chunkBase[b * NC + c] + sx[tid];   // inclusive cumsum of f0_up/FS

    // 2-tap hann-blend upsample coefficients
    int r = t - (t / FR) * FR;
    int fi0 = t / FR;
    int lf0 = fi0 - F0;
    int lf1 = lf0 + 1;
    bool v1 = (fi0 + 1) < T_FRAMES;
    float wA = hannw(FR - r);          // w[80 - r]
    float wB = hannw(2 * FR - r);      // w[160 - r]

    const float* anSb = an_sin + b * T_FRAMES;
    const float* anCb = an_cos + b * T_FRAMES;
    float anS = anSb[fi0] * wA + (v1 ? anSb[fi0 + 1] : 0.0f) * wB;
    float anC = anCb[fi0] * wA + (v1 ? anCb[fi0 + 1] : 0.0f) * wB;

    // phase recurrence: sin/cos(2*pi*n*C) from one sincos
    float theta = TWO_PI * Cval;
    float s1, c1;
    sincosf(theta, &s1, &c1);
    float sn = s1, cnv = c1;
    float numS = 0.0f, denS = 0.0f, numC = 0.0f, denC = 0.0f;

    #pragma unroll 2
    for (int n = 0; n < NHARM; ++n) {
        float bs = s_cnS[lf0 * NHARM + n] * wA + s_cnS[lf1 * NHARM + n] * wB;
        float bc = s_cnC[lf0 * NHARM + n] * wA + s_cnC[lf1 * NHARM + n] * wB;
        bool live = ((float)(n + 1) * f0u) < (FS_F * 0.5f);   // harmonic < Nyquist
        float es = live ? __expf(bs) : 0.0f;                  // TEMP = 1
        float ec = live ? __expf(bc) : 0.0f;
        denS += es; numS = fmaf(es, sn, numS);
        denC += ec; numC = fmaf(ec, cnv, numC);
        float sn2 = fmaf(sn, c1, cnv * s1);                   // angle addition
        float cn2 = fmaf(cnv, c1, -sn * s1);
        sn = sn2; cnv = cn2;
    }
    out[b * T_OUT + t] = anS * (numS / denS) + anC * (numC / denC);
}

extern "C" void kernel_launch(void* const* d_in, const int* in_sizes, int n_in,
                              void* d_out, int out_size, void* d_ws, size_t ws_size,
                              hipStream_t stream) {
    (void)in_sizes; (void)n_in; (void)out_size; (void)ws_size;
    const float* f0     = (const float*)d_in[0];
    const float* cn_sin = (const float*)d_in[1];
    const float* cn_cos = (const float*)d_in[2];
    const float* an_sin = (const float*)d_in[3];
    const float* an_cos = (const float*)d_in[4];
    float* out = (float*)d_out;

    float* chunkSum  = (float*)d_ws;              // [BATCH][NC]
    float* chunkBase = chunkSum + BATCH * NC;     // [BATCH][NC]  (25 KB total)

    chunk_sum_kernel<<<BATCH * NC, CHUNK, 0, stream>>>(f0, chunkSum);
    chunk_scan_kernel<<<BATCH, 512, 0, stream>>>(chunkSum, chunkBase);
    osc_kernel<<<BATCH * NC, CHUNK, 0, stream>>>(f0, cn_sin, cn_cos, an_sin, an_cos,
                                                 chunkBase, out);
}